// GCNLayer_6416681140652
// MI455X (gfx1250) — compile-verified
//
#include <hip/hip_runtime.h>

typedef __attribute__((ext_vector_type(2))) float v2f;
typedef __attribute__((ext_vector_type(8))) float v8f;

#define D_FEAT 128
#define LDS_STRIDE 140   // 140 % 64 = 12 -> conflict-free half-wave bank pattern; 140*4 bytes is 8B-aligned

// ---------------------------------------------------------------------------
// Kernel 1: zero the output accumulator
// ---------------------------------------------------------------------------
__global__ void zero_kernel(float* __restrict__ p, int n) {
    int i = blockIdx.x * blockDim.x + threadIdx.x;
    if (i < n) p[i] = 0.0f;
}

// ---------------------------------------------------------------------------
// Kernel 2: hn = (features @ W^T + b) * norm   via V_WMMA_F32_16X16X4_F32
// Block = 256 threads = 8 waves. Block owns 16 node rows; wave w owns output
// columns [16w, 16w+16). Features tile staged once in LDS, shared by 8 waves.
// ---------------------------------------------------------------------------
__global__ void __launch_bounds__(256)
gemm_wmma_kernel(const float* __restrict__ feat,
                 const float* __restrict__ norm,
                 const float* __restrict__ W,
                 const float* __restrict__ bias,
                 float* __restrict__ hn,
                 int n_nodes) {
    __shared__ float tileA[16 * LDS_STRIDE];

    const int m0        = blockIdx.x * 16;
    const int tid       = threadIdx.x;
    const int wave      = tid >> 5;
    const int lane      = tid & 31;
    const bool fulltile = (m0 + 16 <= n_nodes);

    // ---- stage features[m0 .. m0+16) x 128 into LDS (float4, 2 per thread) ----
    #pragma unroll
    for (int it = 0; it < 2; ++it) {
        int idx = tid + it * 256;        // 0..511 float4 slots
        int r   = idx >> 5;              // row within tile
        int q   = idx & 31;              // float4 within row
        float4 v;
        if (fulltile || (m0 + r < n_nodes)) {
            const float4* row4 = (const float4*)(feat + (size_t)(m0 + r) * D_FEAT);
            v = row4[q];
        } else {
            v = make_float4(0.f, 0.f, 0.f, 0.f);
        }
        float* dstp = &tileA[r * LDS_STRIDE + q * 4];
        dstp[0] = v.x; dstp[1] = v.y; dstp[2] = v.z; dstp[3] = v.w;
    }
    __syncthreads();

    // ---- per-wave WMMA over K ----
    const int n0      = wave * 16;            // output-column tile base
    const int halfsel = lane >> 4;            // 0: K,K+1   1: K+2,K+3  (ISA 16x4 f32 A/B layout)
    const int lrow    = lane & 15;            // M (for A) / N (for B) index within tile

    const float* arow = &tileA[lrow * LDS_STRIDE];               // A: features row
    const float* wrow = W + (size_t)(n0 + lrow) * D_FEAT;        // B: W^T column = W row

    v8f acc = {};
    #pragma unroll 4
    for (int k = 0; k < D_FEAT; k += 4) {
        const int ka = k + halfsel * 2;
        v2f a, b;
        a[0] = arow[ka];  a[1] = arow[ka + 1];
        b[0] = wrow[ka];  b[1] = wrow[ka + 1];
        acc = __builtin_amdgcn_wmma_f32_16x16x4_f32(
                  /*neg_a=*/false, a, /*neg_b=*/false, b,
                  /*c_mod=*/(short)0, acc, /*reuse_a=*/false, /*reuse_b=*/false);
    }

    // ---- epilogue: fold bias + norm, write hn ----
    // D layout: VGPR j -> lanes 0-15: (M=j, N=lane); lanes 16-31: (M=j+8, N=lane-16)
    const float bcol = bias[n0 + lrow];
    const int   rbase = m0 + 8 * halfsel;
    float* outp = hn + (size_t)rbase * D_FEAT + (n0 + lrow);

    if (fulltile) {
        // branch-free fast path: 8 stores, stride D_FEAT
        #pragma unroll
        for (int j = 0; j < 8; ++j) {
            float v = (acc[j] + bcol) * norm[rbase + j];
            outp[(size_t)j * D_FEAT] = v;
        }
    } else {
        #pragma unroll
        for (int j = 0; j < 8; ++j) {
            if (rbase + j < n_nodes) {
                float v = (acc[j] + bcol) * norm[rbase + j];
                outp[(size_t)j * D_FEAT] = v;
            }
        }
    }
}

// ---------------------------------------------------------------------------
// Kernel 3: edge scatter. One wave per edge; lane handles 4 channels.
// agg[dst] += hn[src]  (norm[src] already folded into hn)
// ---------------------------------------------------------------------------
__global__ void __launch_bounds__(256)
edge_scatter_kernel(const float* __restrict__ hn,
                    const int* __restrict__ src,
                    const int* __restrict__ dst,
                    float* __restrict__ agg,
                    int n_edges) {
    int tid  = blockIdx.x * blockDim.x + threadIdx.x;
    int edge = tid >> 5;
    if (edge >= n_edges) return;
    int lane = tid & 31;

    int s = src[edge];
    int d = dst[edge];
    const float* hrow = hn  + (size_t)s * D_FEAT;
    float*       orow = agg + (size_t)d * D_FEAT;

    #pragma unroll
    for (int j = 0; j < 4; ++j) {
        int c = lane + 32 * j;
        atomicAdd(&orow[c], hrow[c]);
    }
}

// ---------------------------------------------------------------------------
// Kernel 4: out = relu(agg * norm)  (in place on d_out)
// ---------------------------------------------------------------------------
__global__ void finalize_kernel(float* __restrict__ out,
                                const float* __restrict__ norm,
                                int total) {
    int i = blockIdx.x * blockDim.x + threadIdx.x;
    if (i < total) {
        float v = out[i] * norm[i >> 7];   // D_FEAT == 128
        out[i] = v > 0.0f ? v : 0.0f;
    }
}

// ---------------------------------------------------------------------------
// Launcher
// ---------------------------------------------------------------------------
extern "C" void kernel_launch(void* const* d_in, const int* in_sizes, int n_in,
                              void* d_out, int out_size, void* d_ws, size_t ws_size,
                              hipStream_t stream) {
    const float* feat = (const float*)d_in[0];   // [N, 128]
    const float* norm = (const float*)d_in[1];   // [N, 1]
    const float* W    = (const float*)d_in[2];   // [128, 128]
    const float* bias = (const float*)d_in[3];   // [128]
    const int*   src  = (const int*)d_in[4];     // [E]
    const int*   dst  = (const int*)d_in[5];     // [E]
    float*       out  = (float*)d_out;           // [N, 128]
    float*       hn   = (float*)d_ws;            // [N, 128] scratch

    const int n_nodes = in_sizes[1];
    const int n_edges = in_sizes[4];
    const int total   = out_size;                // n_nodes * 128

    // 1) zero the accumulator
    zero_kernel<<<(total + 255) / 256, 256, 0, stream>>>(out, total);

    // 2) hn = (feat @ W^T + b) * norm  (WMMA)
    gemm_wmma_kernel<<<(n_nodes + 15) / 16, 256, 0, stream>>>(
        feat, norm, W, bias, hn, n_nodes);

    // 3) scatter-add over edges (one wave per edge)
    {
        long long threads = (long long)n_edges * 32;
        int blocks = (int)((threads + 255) / 256);
        edge_scatter_kernel<<<blocks, 256, 0, stream>>>(hn, src, dst, out, n_edges);
    }

    // 4) out = relu(out * norm)
    finalize_kernel<<<(total + 255) / 256, 256, 0, stream>>>(out, norm, total);
}